// ResNet_15753940042160
// MI455X (gfx1250) — compile-verified
//
#include <hip/hip_runtime.h>
#include <hip/hip_bf16.h>

// ============================================================================
// AdderNet ResNet-18 forward for MI455X (gfx1250, wave32).
//
// Roofline: ~2.4G abs-diff terms, ~11MB weights, <=4MB activations -> fully
// L2-resident (192MB). VALU-bound; WMMA is numerically inapplicable to the
// L1-distance convolutions (|x-w| sums are not bilinear). CDNA5 paths used:
//   * TENSOR_LOAD_TO_LDS (TDM) double-buffered: chunk ch+1's 16x(CB*K*K)
//     weight tile DMA is issued before computing chunk ch; TENSORcnt drained
//     with s_wait_tensorcnt only after the VALU work (true async overlap).
//   * v_wmma_f32_16x16x4_f32 ones-matrix trick for 32-lane reductions in the
//     BN-stats / fc kernels (matrix pipe co-executes with VALU).
//   * global_prefetch (__builtin_prefetch) pulls chunk ch+2's weights to L2.
// NOTE: ping-pong LDS buffers are addressed via runtime offset arithmetic on
// a single __shared__ array; pointer arrays initialized from LDS addresses
// create addrspacecast static initializers that ld.lld rejects.
// ============================================================================

#define EPSV 1e-5f
#define NB 64            // batch

typedef float v2f __attribute__((ext_vector_type(2)));
typedef float v8f __attribute__((ext_vector_type(8)));

// ---------------------------------------------------------------------------
// Cross-lane (32) sum. Primary path: V_WMMA_F32_16X16X4_F32 with B = ones.
// A fragment layout (ISA 7.12.2): lane m<16 holds A[m,0],A[m,1]; lane m+16
// holds A[m,2],A[m,3]. With a = {v,v}:  D[m,n] = 2*(v_m + v_{m+16}).
// Per-lane sum of the 8 D VGPRs gives half-row sums; shfl_xor(16) joins the
// two lane groups -> 2 * total.  Fallback: shuffle tree.
// ---------------------------------------------------------------------------
#if __has_builtin(__builtin_amdgcn_wmma_f32_16x16x4_f32)
#define HAVE_WMMA4 1
#else
#define HAVE_WMMA4 0
#endif

__device__ __forceinline__ float wave_reduce_sum(float v) {
#if HAVE_WMMA4
  v2f a; a.x = v;   a.y = v;
  v2f b; b.x = 1.f; b.y = 1.f;
  v8f c = {};
  v8f d = __builtin_amdgcn_wmma_f32_16x16x4_f32(false, a, false, b,
                                                (short)0, c, false, false);
  float r = d[0] + d[1] + d[2] + d[3] + d[4] + d[5] + d[6] + d[7];
  r += __shfl_xor(r, 16, 32);
  return r * 0.5f;
#else
  #pragma unroll
  for (int off = 16; off > 0; off >>= 1) v += __shfl_xor(v, off, 32);
  return v;
#endif
}

// ---------------------------------------------------------------------------
// TDM 2D tile load: rows x rowlen f32 elements, row stride `rowstride` elts,
// into LDS at byte offset lds_off. D# fields per CDNA5 ISA
// section 8 (group0: count/lds/global/type; group1: data_size, dims, strides).
// ---------------------------------------------------------------------------
#if __has_builtin(__builtin_amdgcn_tensor_load_to_lds)
#define HAVE_TDM 1
typedef unsigned u32x4 __attribute__((ext_vector_type(4)));
typedef int      i32x4 __attribute__((ext_vector_type(4)));
typedef int      i32x8 __attribute__((ext_vector_type(8)));

__device__ __forceinline__ void tdm_load_2d(const void* gsrc, unsigned lds_off,
                                            unsigned rowlen, unsigned rows,
                                            unsigned rowstride) {
  unsigned long long ga = (unsigned long long)(size_t)gsrc;
  u32x4 g0;
  g0[0] = 1u;                                              // count=1 (valid)
  g0[1] = lds_off;                                         // lds_addr bytes
  g0[2] = (unsigned)ga;                                    // global_addr lo
  g0[3] = (unsigned)((ga >> 32) & 0x1ffffffull) | (2u << 30); // addr hi|type=2
  i32x8 g1;
  g1[0] = (int)(2u << 16);                                 // data_size = 4B
  g1[1] = (int)((rowlen & 0xffffu) << 16);                 // tensor_dim0 lo16
  g1[2] = (int)((rows   & 0xffffu) << 16);                 // td0 hi=0 | td1 lo
  g1[3] = (int)((rowlen & 0xffffu) << 16);                 // td1 hi=0 | tile_dim0
  g1[4] = (int)(rows & 0xffffu);                           // tile_dim1 | tile_dim2=0
  g1[5] = (int)rowstride;                                  // dim0_stride lo32
  g1[6] = 0;
  g1[7] = 0;
  i32x4 z4 = {0, 0, 0, 0};
#if __clang_major__ >= 23
  i32x8 z8 = {0, 0, 0, 0, 0, 0, 0, 0};
  __builtin_amdgcn_tensor_load_to_lds(g0, g1, z4, z4, z8, 0);
#else
  __builtin_amdgcn_tensor_load_to_lds(g0, g1, z4, z4, 0);
#endif
}
#else
#define HAVE_TDM 0
#endif

__device__ __forceinline__ void wait_tensor0() {
#if __has_builtin(__builtin_amdgcn_s_wait_tensorcnt)
  __builtin_amdgcn_s_wait_tensorcnt(0);
#endif
}

// ---------------------------------------------------------------------------
// Adder conv:  out[n,o,i,j] = -sum_{c,kh,kw} |xpatch - w|   (zero padding DOES
// contribute |0 - w|, matching conv_general_dilated_patches).
// Block = 256 threads = 8 waves: 16 output channels x 16 pixels.
// Channel-chunked (CB). Software-pipelined: while chunk ch is computed, the
// TDM is DMA-ing chunk ch+1's weight tile and the waves stage chunk ch+1's
// patches into the ping-pong LDS halves; TENSORcnt drained after compute.
// LDS map (floats): [ws0: WE][ws1: WE][xs0: XE][xs1: XE], all offsets from
// the single __shared__ object (LDS byte offset 0).
// ---------------------------------------------------------------------------
template <int K, int CB>
__global__ __launch_bounds__(256)
void adder_conv_kernel(const float* __restrict__ x, const float* __restrict__ w,
                       float* __restrict__ out, int C, int H, int W,
                       int O, int Ho, int Wo, int stride, int pad) {
  constexpr int KK = K * K;
  constexpr int PT = 16;
  constexpr int WE = 16 * CB * KK;   // weight tile elems
  constexpr int XE = CB * PT * KK;   // patch tile elems
  __shared__ float smem[2 * WE + 2 * XE];

  const int tid = threadIdx.x;
  const int ob = blockIdx.x, pt = blockIdx.y, n = blockIdx.z;
  const int o_l = tid >> 4, p_l = tid & 15;
  const int HoWo = Ho * Wo;
  const int pix = pt * PT + p_l;
  const bool pvalid = pix < HoWo;

  const int nch = C / CB;
  const size_t wbase = (size_t)(ob * 16) * C * KK;

  // stage input patches [cl][pixel][tap] for chunk `ch` with zero padding
  auto stage_x = [&](int ch, int bufsel) {
    float* xsm = smem + 2 * WE + bufsel * XE;
    const int c0 = ch * CB;
    for (int i = tid; i < XE; i += 256) {
      int cl = i / (PT * KK);
      int r = i - cl * (PT * KK);
      int p = r / KK, t = r - p * KK;
      int gp = pt * PT + p;
      float v = 0.f;
      if (gp < HoWo) {
        int oh = gp / Wo, ow = gp - oh * Wo;
        int kh = t / K, kw = t - kh * K;
        int ih = oh * stride + kh - pad;
        int iw = ow * stride + kw - pad;
        if (ih >= 0 && ih < H && iw >= 0 && iw < W)
          v = x[(((size_t)n * C + (c0 + cl)) * H + ih) * W + iw];
      }
      xsm[i] = v;
    }
  };
#if !HAVE_TDM
  auto stage_w_sync = [&](const float* wsrc, int bufsel) {
    float* wsm = smem + bufsel * WE;
    for (int i = tid; i < WE; i += 256) {
      int ol = i / (CB * KK), r = i - ol * (CB * KK);
      wsm[i] = wsrc[(size_t)ol * C * KK + r];
    }
  };
#endif

  // ---- prologue: stage chunk 0 into buffer half 0 ----
  {
    const float* wsrc = w + wbase;
#if HAVE_TDM
    if (tid == 0)   // one wave issues the DMA (EXEC ignored by TDM)
      tdm_load_2d(wsrc, 0u, (unsigned)(CB * KK), 16u, (unsigned)(C * KK));
#else
    stage_w_sync(wsrc, 0);
#endif
    stage_x(0, 0);
#if HAVE_TDM
    if (tid < 32) wait_tensor0();
#endif
    __syncthreads();
  }

  float acc = 0.f;
  for (int ch = 0; ch < nch; ++ch) {
    const int cur = ch & 1, nxt = cur ^ 1;
    const bool have_next = (ch + 1 < nch);

    // ---- kick off chunk ch+1 into the other LDS half (no wait yet) ----
    if (have_next) {
      const float* wsrc_n = w + wbase + (size_t)(ch + 1) * CB * KK;
#if HAVE_TDM
      if (tid == 0)
        tdm_load_2d(wsrc_n, (unsigned)(nxt * WE * 4), (unsigned)(CB * KK), 16u,
                    (unsigned)(C * KK));
#else
      stage_w_sync(wsrc_n, nxt);
#endif
      stage_x(ch + 1, nxt);
      if (ch + 2 < nch)   // warm L2 for the DMA after that
        __builtin_prefetch(wsrc_n + CB * KK, 0, 1);
    }

    // ---- compute chunk ch (overlaps with the in-flight DMA) ----
    if (pvalid) {
      const float* xbase = smem + 2 * WE + cur * XE;
      const float* wbaseL = smem + cur * WE;
      #pragma unroll
      for (int cl = 0; cl < CB; ++cl) {
        const float* xp = &xbase[(cl * PT + p_l) * KK];
        const float* wp = &wbaseL[(o_l * CB + cl) * KK];
        #pragma unroll
        for (int t = 0; t < KK; ++t) acc += fabsf(xp[t] - wp[t]);
      }
    }

#if HAVE_TDM
    if (have_next && tid < 32) wait_tensor0();  // drain this iteration's DMA
#endif
    __syncthreads();  // next iteration may read `nxt` / overwrite `cur`
  }
  if (pvalid) out[((size_t)n * O + ob * 16 + o_l) * HoWo + pix] = -acc;
}

// ---------------------------------------------------------------------------
// BN batch statistics over (N,H,W): mean + rsqrt(var+eps). One block/channel.
// ---------------------------------------------------------------------------
__global__ __launch_bounds__(256)
void bn_stats_kernel(const float* __restrict__ x, float* __restrict__ mean,
                     float* __restrict__ istd, int C, int HW, int N, float eps) {
  const int c = blockIdx.x, tid = threadIdx.x;
  const int total = N * HW;
  float s = 0.f, q = 0.f;
  for (int i = tid; i < total; i += 256) {
    int n = i / HW, r = i - n * HW;
    float v = x[((size_t)n * C + c) * HW + r];
    s += v;
    q += v * v;
  }
  __shared__ float ss[256], sq[256];
  ss[tid] = s; sq[tid] = q;
  __syncthreads();
  if (tid < 32) {   // wave0, EXEC all ones -> WMMA legal
    float a = 0.f, b = 0.f;
    #pragma unroll
    for (int j = 0; j < 8; ++j) { a += ss[tid + 32 * j]; b += sq[tid + 32 * j]; }
    a = wave_reduce_sum(a);
    b = wave_reduce_sum(b);
    if (tid == 0) {
      float inv = 1.f / (float)total;
      float m = a * inv;
      float var = b * inv - m * m;
      mean[c] = m;
      istd[c] = rsqrtf(var + eps);
    }
  }
}

// mode: bit0 = ReLU, bit1 = add residual
__global__ __launch_bounds__(256)
void bn_apply_kernel(const float* x, const float* res,
                     const float* __restrict__ g, const float* __restrict__ b,
                     const float* __restrict__ mean, const float* __restrict__ istd,
                     float* out, int C, int HW, int total, int mode) {
  int i = blockIdx.x * 256 + threadIdx.x;
  if (i >= total) return;
  int c = (i / HW) % C;
  float v = (x[i] - mean[c]) * istd[c] * g[c] + b[c];
  if (mode & 2) v += res[i];
  if (mode & 1) v = fmaxf(v, 0.f);
  out[i] = v;
}

// maxpool 3x3 stride 2 pad 1 (padding = -inf identity)
__global__ __launch_bounds__(256)
void maxpool_kernel(const float* __restrict__ x, float* __restrict__ out,
                    int C, int H, int W, int Ho, int Wo, int total) {
  int i = blockIdx.x * 256 + threadIdx.x;
  if (i >= total) return;
  int wo = i % Wo; int t = i / Wo;
  int ho = t % Ho; t /= Ho;
  int c = t % C;   int n = t / C;
  float m = -3.402823466e38f;
  #pragma unroll
  for (int kh = 0; kh < 3; ++kh) {
    int ih = ho * 2 + kh - 1;
    if (ih < 0 || ih >= H) continue;
    #pragma unroll
    for (int kw = 0; kw < 3; ++kw) {
      int iw = wo * 2 + kw - 1;
      if (iw < 0 || iw >= W) continue;
      m = fmaxf(m, x[(((size_t)n * C + c) * H + ih) * W + iw]);
    }
  }
  out[i] = m;
}

// adder fc: out[n,o] = -sum_c |h[n,c] - w[o,c]|
__global__ __launch_bounds__(128)
void fc_adder_kernel(const float* __restrict__ h, const float* __restrict__ w,
                     float* __restrict__ out, int C, int O) {
  const int n = blockIdx.x, o = blockIdx.y, tid = threadIdx.x;
  float s = 0.f;
  for (int c = tid; c < C; c += 128)
    s += fabsf(h[(size_t)n * C + c] - w[(size_t)o * C + c]);
  __shared__ float red[128];
  red[tid] = s;
  __syncthreads();
  if (tid < 32) {
    float v = red[tid] + red[tid + 32] + red[tid + 64] + red[tid + 96];
    v = wave_reduce_sum(v);
    if (tid == 0) out[(size_t)n * O + o] = -v;
  }
}

// ============================================================================
// Host side
// ============================================================================
struct BlockP {
  const float *w1, *g1, *b1, *w2, *g2, *b2, *wd, *gd, *bd;
};

static void run_conv(int K, const float* x, const float* w, float* out,
                     int C, int H, int W, int O, int Ho, int Wo,
                     int stride, int pad, hipStream_t st) {
  dim3 grid(O / 16, (Ho * Wo + 15) / 16, NB);
  if (K == 7)
    adder_conv_kernel<7, 3><<<grid, 256, 0, st>>>(x, w, out, C, H, W, O, Ho, Wo, stride, pad);
  else if (K == 3)
    adder_conv_kernel<3, 8><<<grid, 256, 0, st>>>(x, w, out, C, H, W, O, Ho, Wo, stride, pad);
  else
    adder_conv_kernel<1, 32><<<grid, 256, 0, st>>>(x, w, out, C, H, W, O, Ho, Wo, stride, pad);
}

extern "C" void kernel_launch(void* const* d_in, const int* in_sizes, int n_in,
                              void* d_out, int out_size, void* d_ws, size_t ws_size,
                              hipStream_t stream) {
  (void)out_size; (void)ws_size; (void)n_in;
  // ---- input mapping: handle both insertion-order and jax-sorted flatten ----
  const float *x, *conv1_w, *bn1_g, *bn1_b, *fc_w, *bn2_g, *bn2_b;
  BlockP blocks[4][2];
  const bool hasdown[4] = {false, true, true, true};
  auto F = [&](int idx) { return (const float*)d_in[idx]; };
  int i = 0;
  if (in_sizes[0] == 64 * 3 * 32 * 32) {
    // ordering A: dict insertion order, depth-first
    x = F(i++); conv1_w = F(i++); bn1_g = F(i++); bn1_b = F(i++);
    for (int s = 0; s < 4; ++s)
      for (int bl = 0; bl < 2; ++bl) {
        BlockP& p = blocks[s][bl];
        p.w1 = F(i++); p.g1 = F(i++); p.b1 = F(i++);
        p.w2 = F(i++); p.g2 = F(i++); p.b2 = F(i++);
        if (bl == 0 && hasdown[s]) { p.wd = F(i++); p.gd = F(i++); p.bd = F(i++); }
        else { p.wd = p.gd = p.bd = nullptr; }
      }
    fc_w = F(i++); bn2_g = F(i++); bn2_b = F(i++);
  } else {
    // ordering B: jax.tree_util flatten (dict keys sorted), params < x
    bn1_b = F(i++); bn1_g = F(i++); bn2_b = F(i++); bn2_g = F(i++);
    conv1_w = F(i++); fc_w = F(i++);
    for (int s = 0; s < 4; ++s)
      for (int bl = 0; bl < 2; ++bl) {
        BlockP& p = blocks[s][bl];
        if (bl == 0 && hasdown[s]) {
          p.b1 = F(i++); p.b2 = F(i++); p.bd = F(i++);
          p.g1 = F(i++); p.g2 = F(i++); p.gd = F(i++);
          p.w1 = F(i++); p.w2 = F(i++); p.wd = F(i++);
        } else {
          p.b1 = F(i++); p.b2 = F(i++); p.g1 = F(i++); p.g2 = F(i++);
          p.w1 = F(i++); p.w2 = F(i++);
          p.wd = p.gd = p.bd = nullptr;
        }
      }
    x = F(i++);
  }

  // ---- workspace layout ----
  float* wsf = (float*)d_ws;
  float* buf[4] = {wsf, wsf + 1048576, wsf + 2 * 1048576, wsf + 3 * 1048576};
  float* stA_m = wsf + 4 * 1048576;  // 512
  float* stA_s = stA_m + 512;
  float* stB_m = stA_s + 512;
  float* stB_s = stB_m + 512;
  float* fcbuf = stB_s + 512;        // 640

  // ---- conv1: [64,3,32,32] -> [64,64,16,16], BN+ReLU, maxpool -> [64,64,8,8]
  run_conv(7, x, conv1_w, buf[1], 3, 32, 32, 64, 16, 16, 2, 3, stream);
  bn_stats_kernel<<<64, 256, 0, stream>>>(buf[1], stA_m, stA_s, 64, 256, NB, EPSV);
  {
    int tot = NB * 64 * 256;
    bn_apply_kernel<<<(tot + 255) / 256, 256, 0, stream>>>(
        buf[1], buf[1], bn1_g, bn1_b, stA_m, stA_s, buf[1], 64, 256, tot, 1);
  }
  {
    int tot = NB * 64 * 64;
    maxpool_kernel<<<(tot + 255) / 256, 256, 0, stream>>>(buf[1], buf[0], 64, 16, 16, 8, 8, tot);
  }

  // ---- stages ----
  const int planes[4] = {64, 128, 256, 512};
  const int sstride[4] = {1, 2, 2, 2};
  int hi = 0, Cin = 64, Hs = 8;
  for (int s = 0; s < 4; ++s) {
    for (int bl = 0; bl < 2; ++bl) {
      const BlockP& p = blocks[s][bl];
      int stride = (bl == 0) ? sstride[s] : 1;
      int pl = planes[s];
      int Hout = (Hs - 1) / stride + 1;
      int HWo = Hout * Hout;
      int tot = NB * pl * HWo;
      int f[3], k = 0;
      for (int q = 0; q < 4; ++q)
        if (q != hi) f[k++] = q;
      float* t1 = buf[f[0]];
      float* t2 = buf[f[1]];
      float* t3 = buf[f[2]];

      run_conv(3, buf[hi], p.w1, t1, Cin, Hs, Hs, pl, Hout, Hout, stride, 1, stream);
      bn_stats_kernel<<<pl, 256, 0, stream>>>(t1, stA_m, stA_s, pl, HWo, NB, EPSV);
      bn_apply_kernel<<<(tot + 255) / 256, 256, 0, stream>>>(
          t1, t1, p.g1, p.b1, stA_m, stA_s, t1, pl, HWo, tot, 1);

      run_conv(3, t1, p.w2, t2, pl, Hout, Hout, pl, Hout, Hout, 1, 1, stream);
      bn_stats_kernel<<<pl, 256, 0, stream>>>(t2, stA_m, stA_s, pl, HWo, NB, EPSV);

      const float* idn;
      if (p.wd) {
        run_conv(1, buf[hi], p.wd, t3, Cin, Hs, Hs, pl, Hout, Hout, stride, 0, stream);
        bn_stats_kernel<<<pl, 256, 0, stream>>>(t3, stB_m, stB_s, pl, HWo, NB, EPSV);
        bn_apply_kernel<<<(tot + 255) / 256, 256, 0, stream>>>(
            t3, t3, p.gd, p.bd, stB_m, stB_s, t3, pl, HWo, tot, 0);
        idn = t3;
      } else {
        idn = buf[hi];
      }
      // out = relu(bn(conv2) + idn), in-place into t2
      bn_apply_kernel<<<(tot + 255) / 256, 256, 0, stream>>>(
          t2, idn, p.g2, p.b2, stA_m, stA_s, t2, pl, HWo, tot, 3);

      hi = f[1];
      Cin = pl;
      Hs = Hout;
    }
  }

  // ---- head: stage4 output is [64,512,1,1] -> avgpool is identity ----
  fc_adder_kernel<<<dim3(NB, 10), 128, 0, stream>>>(buf[hi], fc_w, fcbuf, 512, 10);
  bn_stats_kernel<<<10, 256, 0, stream>>>(fcbuf, stB_m, stB_s, 10, 1, NB, EPSV);
  bn_apply_kernel<<<(640 + 255) / 256, 256, 0, stream>>>(
      fcbuf, fcbuf, bn2_g, bn2_b, stB_m, stB_s, (float*)d_out, 10, 1, 640, 0);
}